// SSD_37443524887267
// MI455X (gfx1250) — compile-verified
//
#include <hip/hip_runtime.h>
#include <math.h>

#define NUM_CLASSES 21
#define TOP_K       200
#define CONF_THRESH 0.01f
#define NMS_THRESH  0.045f
#define VAR0        0.1f
#define VAR1        0.2f
#define BATCH       128
#define NANCH       8732
#define KP          208   // TOP_K padded to multiple of 16 for WMMA tiling
#define NT          (KP / 16)   // 13 tiles per dimension

typedef __attribute__((ext_vector_type(2))) float v2f;
typedef __attribute__((ext_vector_type(8))) float v8f;

// ---------------------------------------------------------------------------
// Kernel 1: per-anchor softmax stats (row max and 1/sum of exp). Pure
// bandwidth: reads 94MB of conf once, writes 8.9MB of stats.
// ---------------------------------------------------------------------------
__global__ void SSD_softmax_stats(const float* __restrict__ conf,
                                  float* __restrict__ d_max,
                                  float* __restrict__ d_invsum,
                                  int total) {
    int g = blockIdx.x * blockDim.x + threadIdx.x;
    if (g >= total) return;
    const float* row = conf + (size_t)g * NUM_CLASSES;
    float m = row[0];
#pragma unroll
    for (int c = 1; c < NUM_CLASSES; ++c) m = fmaxf(m, row[c]);
    float s = 0.0f;
#pragma unroll
    for (int c = 0; c < NUM_CLASSES; ++c) s += __expf(row[c] - m);
    d_max[g]    = m;
    d_invsum[g] = 1.0f / s;
}

// ---------------------------------------------------------------------------
// Kernel 2: per (batch, class) block. Build masked softmax scores for one
// class in LDS, run 200 rounds of block-wide argmax (strict-greater keeps
// lowest index first, matching jax.lax.top_k tie-break), then decode only the
// 200 selected boxes and stage (score, x1,y1,x2,y2) into d_out.
// ---------------------------------------------------------------------------
__global__ void SSD_topk_decode(const float* __restrict__ loc,
                                const float* __restrict__ conf,
                                const float* __restrict__ anchors,
                                const float* __restrict__ d_max,
                                const float* __restrict__ d_invsum,
                                float* __restrict__ out) {
    const int bc  = blockIdx.x;
    const int b   = bc / NUM_CLASSES;
    const int c   = bc % NUM_CLASSES;
    const int tid = threadIdx.x;
    if (c == 0) return;  // background: NMS kernel zero-fills it

    __shared__ float s_sc[NANCH];
    __shared__ float r_val[256];
    __shared__ int   r_idx[256];
    __shared__ float s_val[TOP_K];
    __shared__ int   s_idx[TOP_K];

    const size_t rowbase = (size_t)b * NANCH;

    // Masked class scores into LDS (conf fits in 192MB L2 -> re-reads on-chip)
    for (int n = tid; n < NANCH; n += 256) {
        const float* row = conf + (rowbase + n) * NUM_CLASSES;
        float m, isum;
        if (d_max != nullptr) {
            m    = d_max[rowbase + n];
            isum = d_invsum[rowbase + n];
        } else {  // fallback when workspace is too small: recompute inline
            m = row[0];
#pragma unroll
            for (int k = 1; k < NUM_CLASSES; ++k) m = fmaxf(m, row[k]);
            float s = 0.0f;
#pragma unroll
            for (int k = 0; k < NUM_CLASSES; ++k) s += __expf(row[k] - m);
            isum = 1.0f / s;
        }
        float p = __expf(row[c] - m) * isum;
        s_sc[n] = (p > CONF_THRESH) ? p : 0.0f;
    }
    __syncthreads();

    // 200 rounds of block-wide argmax with lowest-index tie-break.
    for (int k = 0; k < TOP_K; ++k) {
        float best = -1.0f;
        int   bi   = NANCH;
        for (int n = tid; n < NANCH; n += 256) {
            float v = s_sc[n];
            if (v > best) { best = v; bi = n; }  // ascending n: first occurrence wins
        }
        r_val[tid] = best;
        r_idx[tid] = bi;
        __syncthreads();
        for (int s = 128; s > 0; s >>= 1) {
            if (tid < s) {
                float v = r_val[tid + s];
                int   ix = r_idx[tid + s];
                if (v > r_val[tid] || (v == r_val[tid] && ix < r_idx[tid])) {
                    r_val[tid] = v;
                    r_idx[tid] = ix;
                }
            }
            __syncthreads();
        }
        if (tid == 0) {
            s_val[k] = r_val[0];
            s_idx[k] = r_idx[0];
            s_sc[r_idx[0]] = -1.0f;  // below any masked score (>= 0)
        }
        __syncthreads();
    }

    // Decode only the selected 200 boxes; stage into output layout.
    float* obase = out + (size_t)bc * TOP_K * 5;
    for (int k = tid; k < TOP_K; k += 256) {
        int n = s_idx[k];
        const float* lp = loc + (rowbase + n) * 4;
        const float* ap = anchors + (size_t)n * 4;
        float l0 = lp[0], l1 = lp[1], l2 = lp[2], l3 = lp[3];
        float a0 = ap[0], a1 = ap[1], a2 = ap[2], a3 = ap[3];
        float cx = a0 + l0 * VAR0 * a2;
        float cy = a1 + l1 * VAR0 * a3;
        float w  = a2 * __expf(l2 * VAR1);
        float h  = a3 * __expf(l3 * VAR1);
        obase[k * 5 + 0] = s_val[k];
        obase[k * 5 + 1] = cx - 0.5f * w;
        obase[k * 5 + 2] = cy - 0.5f * h;
        obase[k * 5 + 3] = cx + 0.5f * w;
        obase[k * 5 + 4] = cy + 0.5f * h;
    }
}

// ---------------------------------------------------------------------------
// Kernel 3: per (batch, class) block NMS + stable pack.
// The pairwise union term (area_i + area_j + eps) over each 16x16 tile is a
// rank-2 f32 matmul computed with V_WMMA_F32_16X16X4_F32:
//   A = [area_m, 1, 0, 0] (16x4), B = [1; area_n; 0; 0] (4x16), C = eps.
// One wave32 per tile; intersection term is VALU min/max on LDS boxes.
// The threshold test uses multiply-compare (denominator is strictly
// positive), avoiding the full IEEE f32 division expansion per pair:
//   inter/(sum - inter) > T  <=>  inter > T*(sum - inter)
// ---------------------------------------------------------------------------
__global__ void SSD_nms_pack(float* __restrict__ out) {
    const int bc  = blockIdx.x;
    const int c   = bc % NUM_CLASSES;
    const int tid = threadIdx.x;
    float* obase  = out + (size_t)bc * TOP_K * 5;

    if (c == 0) {  // background class: all zeros
        for (int k = tid; k < TOP_K * 5; k += 256) obase[k] = 0.0f;
        return;
    }

    __shared__ float s_score[KP];
    __shared__ float s_x1[KP], s_y1[KP], s_x2[KP], s_y2[KP];
    __shared__ float s_area[KP];
    __shared__ unsigned char s_sup[KP * KP];  // sup[m][n] = iou(m,n) > thresh
    __shared__ unsigned char s_keep[KP];
    __shared__ short s_pos[KP];

    // Load staged top-K entries; pad rows 200..207 with zeros.
    for (int k = tid; k < KP; k += 256) {
        if (k < TOP_K) {
            float sc = obase[k * 5 + 0];
            float x1 = obase[k * 5 + 1];
            float y1 = obase[k * 5 + 2];
            float x2 = obase[k * 5 + 3];
            float y2 = obase[k * 5 + 4];
            s_score[k] = sc;
            s_x1[k] = x1; s_y1[k] = y1; s_x2[k] = x2; s_y2[k] = y2;
            s_area[k]  = (x2 - x1) * (y2 - y1);
            s_keep[k]  = (sc > CONF_THRESH) ? 1 : 0;
        } else {
            s_score[k] = 0.0f;
            s_x1[k] = 0.0f; s_y1[k] = 0.0f; s_x2[k] = 0.0f; s_y2[k] = 0.0f;
            s_area[k]  = 0.0f;
            s_keep[k]  = 0;
        }
    }
    __syncthreads();

    // 13x13 = 169 tiles of 16x16; 8 wave32s round-robin over tiles.
    const int wave = tid >> 5;
    const int lane = tid & 31;
    for (int t = wave; t < NT * NT; t += 8) {
        const int tm = t / NT;
        const int tn = t % NT;

        // A-matrix 16x4 layout: lanes 0-15 hold K={0,1}; lanes 16-31 K={2,3}=0.
        v2f a, bmat;
        if (lane < 16) {
            a.x    = s_area[tm * 16 + lane];  // K=0: area of row m
            a.y    = 1.0f;                    // K=1: ones
            bmat.x = 1.0f;                    // B[0][n] = 1
            bmat.y = s_area[tn * 16 + lane];  // B[1][n] = area of col n
        } else {
            a.x = 0.0f; a.y = 0.0f;
            bmat.x = 0.0f; bmat.y = 0.0f;
        }
        v8f cc;
#pragma unroll
        for (int r = 0; r < 8; ++r) cc[r] = 1e-12f;  // eps folded into C

        // D[m][n] = area_m + area_n + 1e-12 (exact f32 rank-2 matmul)
        v8f d = __builtin_amdgcn_wmma_f32_16x16x4_f32(
            false, a, false, bmat, (short)0, cc, false, false);

        // D layout: VGPR r -> row r (lanes 0-15) / row r+8 (lanes 16-31)
        const int nloc  = lane & 15;
        const int mbase = (lane < 16) ? 0 : 8;
#pragma unroll
        for (int r = 0; r < 8; ++r) {
            const int m = tm * 16 + mbase + r;
            const int n = tn * 16 + nloc;
            float ltx = fmaxf(s_x1[m], s_x1[n]);
            float lty = fmaxf(s_y1[m], s_y1[n]);
            float rbx = fminf(s_x2[m], s_x2[n]);
            float rby = fminf(s_y2[m], s_y2[n]);
            float w = fmaxf(rbx - ltx, 0.0f);
            float h = fmaxf(rby - lty, 0.0f);
            float inter = w * h;
            // iou > T  <=>  inter > T * (union)   [union > 0 always]
            float unin = d[r] - inter;  // area_m + area_n - inter + 1e-12
            s_sup[m * KP + n] = (inter > NMS_THRESH * unin) ? 1 : 0;
        }
    }
    __syncthreads();

    // Greedy sequential suppression (inner clear parallel across threads).
    for (int i = 0; i < TOP_K; ++i) {
        if (s_keep[i]) {
            const unsigned char* row = &s_sup[i * KP];
            for (int j = i + 1 + tid; j < TOP_K; j += 256)
                if (row[j]) s_keep[j] = 0;
        }
        __syncthreads();
    }

    // Stable pack positions (original order == descending score order).
    if (tid == 0) {
        int p = 0;
        for (int i = 0; i < TOP_K; ++i)
            s_pos[i] = s_keep[i] ? (short)(p++) : (short)-1;
    }
    __syncthreads();

    // Zero-fill output, then scatter kept entries to packed positions.
    for (int k = tid; k < TOP_K * 5; k += 256) obase[k] = 0.0f;
    __syncthreads();
    for (int i = tid; i < TOP_K; i += 256) {
        int p = s_pos[i];
        if (p >= 0) {
            obase[p * 5 + 0] = s_score[i];
            obase[p * 5 + 1] = s_x1[i];
            obase[p * 5 + 2] = s_y1[i];
            obase[p * 5 + 3] = s_x2[i];
            obase[p * 5 + 4] = s_y2[i];
        }
    }
}

// ---------------------------------------------------------------------------
extern "C" void kernel_launch(void* const* d_in, const int* in_sizes, int n_in,
                              void* d_out, int out_size, void* d_ws, size_t ws_size,
                              hipStream_t stream) {
    (void)in_sizes; (void)n_in; (void)out_size;
    const float* loc     = (const float*)d_in[0];
    const float* conf    = (const float*)d_in[1];
    const float* anchors = (const float*)d_in[2];
    float* out = (float*)d_out;

    const int    total    = BATCH * NANCH;
    const size_t ws_need  = (size_t)2 * total * sizeof(float);
    float* wmax = nullptr;
    float* winv = nullptr;
    if (d_ws != nullptr && ws_size >= ws_need) {
        wmax = (float*)d_ws;
        winv = wmax + total;
        SSD_softmax_stats<<<(total + 255) / 256, 256, 0, stream>>>(conf, wmax, winv, total);
    }
    SSD_topk_decode<<<BATCH * NUM_CLASSES, 256, 0, stream>>>(loc, conf, anchors, wmax, winv, out);
    SSD_nms_pack<<<BATCH * NUM_CLASSES, 256, 0, stream>>>(out);
}